// _SingleFMCTracker_72593537237542
// MI455X (gfx1250) — compile-verified
//
#include <hip/hip_runtime.h>
#include <hip/hip_bf16.h>
#include <math.h>

// ---------------------------------------------------------------------------
// Problem constants (from reference): x [B=64, C=3, H=512, W=512] f32
// out [B, 2] f32 : sigmoid(stack([col_out, row_out], axis=1))
// ---------------------------------------------------------------------------
#define NN   512
#define HW   (NN * NN)
#define NB   64
#define NC   3
#define WGS_PER_BATCH 16

typedef float v2f __attribute__((ext_vector_type(2)));
typedef float v8f __attribute__((ext_vector_type(8)));

// Workspace layout (in floats)
#define OFF_D     0
#define OFF_WROW  (OFF_D    + HW)
#define OFF_WCOL  (OFF_WROW + HW)
#define OFF_PROW  (OFF_WCOL + HW)
#define OFF_PCOL  (OFF_PROW + HW)
#define OFF_EROW  (OFF_PCOL + HW)
#define OFF_ECOL  (OFF_EROW + HW)
#define OFF_PART  (OFF_ECOL + HW)   // NB * WGS_PER_BATCH * 2 floats

// ---------------------------------------------------------------------------
// Build orthonormal DCT-II matrix D[k*N + n] = s_k * cos(pi*(n+0.5)*k/N)
// ---------------------------------------------------------------------------
__global__ __launch_bounds__(256)
void build_dct_kernel(float* __restrict__ D) {
    int i = blockIdx.x * blockDim.x + threadIdx.x;
    if (i >= HW) return;
    int k = i >> 9;          // i / 512
    int n = i & (NN - 1);    // i % 512
    const float pi = 3.14159265358979323846f;
    float s = (k == 0) ? sqrtf(1.0f / NN) : sqrtf(2.0f / NN);
    float ang = pi * ((float)n + 0.5f) * (float)k / (float)NN;
    D[i] = s * __cosf(ang) == 0.0f && false ? 0.0f : s * cosf(ang);
}

// ---------------------------------------------------------------------------
// W_row[k,l] = row_w[k]*att[k,l]; W_col[k,l] = col_w[l]*att[k,l]
// ---------------------------------------------------------------------------
__global__ __launch_bounds__(256)
void build_weights_kernel(const float* __restrict__ att,
                          const float* __restrict__ row_w,
                          const float* __restrict__ col_w,
                          float* __restrict__ Wrow,
                          float* __restrict__ Wcol) {
    int i = blockIdx.x * blockDim.x + threadIdx.x;
    if (i >= HW) return;
    int k = i >> 9;
    int l = i & (NN - 1);
    float a = att[i];
    Wrow[i] = row_w[k] * a;
    Wcol[i] = col_w[l] * a;
}

// ---------------------------------------------------------------------------
// 512x512x512 f32 GEMM using V_WMMA_F32_16X16X4_F32 (wave32, 1 wave/block,
// one 16x16 C tile per wave, K stepped by 4).
//   C = op(A) * B,  op = transpose if TRANSA
// Fragment mapping (ISA 7.12.2):
//   A (16x4 f32, 2 VGPR): lane<16 -> K={0,1}; lane>=16 -> K={2,3}; M = lane%16
//   B (4x16 f32, 2 VGPR): row-striped: K = vgpr + 2*(lane/16), N = lane%16
//   C/D (16x16 f32, 8 VGPR): row r -> M = r + 8*(lane/16), N = lane%16
// ---------------------------------------------------------------------------
template <bool TRANSA>
__global__ __launch_bounds__(32)
void gemm512_wmma_kernel(const float* __restrict__ A,
                         const float* __restrict__ B,
                         float* __restrict__ C) {
    const int n0   = blockIdx.x * 16;
    const int m0   = blockIdx.y * 16;
    const int lane = threadIdx.x;
    const int half = lane >> 4;     // 0 or 1
    const int l16  = lane & 15;

    v8f acc = {};
    for (int k0 = 0; k0 < NN; k0 += 4) {
        const int ka = k0 + 2 * half;
        v2f a, b;
        if (TRANSA) {
            a.x = A[(size_t)(ka + 0) * NN + (m0 + l16)];
            a.y = A[(size_t)(ka + 1) * NN + (m0 + l16)];
        } else {
            a.x = A[(size_t)(m0 + l16) * NN + ka + 0];
            a.y = A[(size_t)(m0 + l16) * NN + ka + 1];
        }
        b.x = B[(size_t)(ka + 0) * NN + n0 + l16];
        b.y = B[(size_t)(ka + 1) * NN + n0 + l16];
        // D = A x B + C   (8-arg form: neg_a, A, neg_b, B, c_mod, C, reuse_a, reuse_b)
        acc = __builtin_amdgcn_wmma_f32_16x16x4_f32(
            false, a, false, b, (short)0, acc, false, false);
    }
#pragma unroll
    for (int r = 0; r < 8; ++r) {
        C[(size_t)(m0 + r + 8 * half) * NN + n0 + l16] = acc[r];
    }
}

// ---------------------------------------------------------------------------
// Streaming reduction: for each batch b,
//   r_acc = sum_{h,w} Erow[h,w] * (x[b,0,h,w]+x[b,1,h,w]+x[b,2,h,w])
//   c_acc = same with Ecol
// float4 loads; Erow/Ecol are L2-resident (2 MiB) across all 64 batches.
// Two-stage deterministic reduction: per-WG partials -> finalize kernel.
// ---------------------------------------------------------------------------
__global__ __launch_bounds__(256)
void batch_reduce_kernel(const float* __restrict__ x,
                         const float* __restrict__ Erow,
                         const float* __restrict__ Ecol,
                         float* __restrict__ partials) {
    const int b  = blockIdx.y;
    const int wg = blockIdx.x;
    const int nvec = HW / 4;                       // 65536 float4 per plane
    const float4* __restrict__ er4 = (const float4*)Erow;
    const float4* __restrict__ ec4 = (const float4*)Ecol;
    const float4* __restrict__ x4  =
        (const float4*)(x + (size_t)b * NC * HW);

    float r_acc = 0.0f, c_acc = 0.0f;
    const int stride = WGS_PER_BATCH * 256;
    for (int i = wg * 256 + threadIdx.x; i < nvec; i += stride) {
        // prefetch next strip of x (emits global_prefetch_b8)
        __builtin_prefetch(&x4[i + stride], 0, 1);
        float4 er = er4[i];
        float4 ec = ec4[i];
        float4 x0 = x4[i];
        float4 x1 = x4[i + nvec];
        float4 x2 = x4[i + 2 * nvec];
        float sx = x0.x + x1.x + x2.x;
        float sy = x0.y + x1.y + x2.y;
        float sz = x0.z + x1.z + x2.z;
        float sw = x0.w + x1.w + x2.w;
        r_acc += sx * er.x + sy * er.y + sz * er.z + sw * er.w;
        c_acc += sx * ec.x + sy * ec.y + sz * ec.z + sw * ec.w;
    }

    __shared__ float sr[256];
    __shared__ float sc[256];
    sr[threadIdx.x] = r_acc;
    sc[threadIdx.x] = c_acc;
    __syncthreads();
    for (int off = 128; off > 0; off >>= 1) {
        if (threadIdx.x < off) {
            sr[threadIdx.x] += sr[threadIdx.x + off];
            sc[threadIdx.x] += sc[threadIdx.x + off];
        }
        __syncthreads();
    }
    if (threadIdx.x == 0) {
        partials[(size_t)(b * WGS_PER_BATCH + wg) * 2 + 0] = sr[0];
        partials[(size_t)(b * WGS_PER_BATCH + wg) * 2 + 1] = sc[0];
    }
}

// ---------------------------------------------------------------------------
// Finalize: sum WG partials per batch, scale by 1/(C*H*W), sigmoid,
// store out[b] = [sigmoid(col), sigmoid(row)]  (reference stacks col first).
// ---------------------------------------------------------------------------
__global__ __launch_bounds__(32)
void finalize_kernel(const float* __restrict__ partials,
                     float* __restrict__ out) {
    const int b = blockIdx.x;
    float r = 0.0f, c = 0.0f;
    if ((int)threadIdx.x < WGS_PER_BATCH) {
        r = partials[(size_t)(b * WGS_PER_BATCH + threadIdx.x) * 2 + 0];
        c = partials[(size_t)(b * WGS_PER_BATCH + threadIdx.x) * 2 + 1];
    }
#pragma unroll
    for (int off = 16; off > 0; off >>= 1) {
        r += __shfl_down(r, off, 32);
        c += __shfl_down(c, off, 32);
    }
    if (threadIdx.x == 0) {
        const float scale = 1.0f / ((float)NC * (float)NN * (float)NN);
        float ro = r * scale;
        float co = c * scale;
        out[(size_t)b * 2 + 0] = 1.0f / (1.0f + expf(-co));
        out[(size_t)b * 2 + 1] = 1.0f / (1.0f + expf(-ro));
    }
}

// ---------------------------------------------------------------------------
extern "C" void kernel_launch(void* const* d_in, const int* in_sizes, int n_in,
                              void* d_out, int out_size, void* d_ws, size_t ws_size,
                              hipStream_t stream) {
    const float* x     = (const float*)d_in[0];  // [64,3,512,512]
    const float* att   = (const float*)d_in[1];  // [512,512]
    const float* row_w = (const float*)d_in[2];  // [512]
    const float* col_w = (const float*)d_in[3];  // [512]
    float* out = (float*)d_out;                  // [64,2]
    float* ws  = (float*)d_ws;

    float* D    = ws + OFF_D;
    float* Wrow = ws + OFF_WROW;
    float* Wcol = ws + OFF_WCOL;
    float* Prow = ws + OFF_PROW;
    float* Pcol = ws + OFF_PCOL;
    float* Erow = ws + OFF_EROW;
    float* Ecol = ws + OFF_ECOL;
    float* part = ws + OFF_PART;

    const int threads = 256;
    const int nblk = (HW + threads - 1) / threads;

    // 1) DCT basis + weight matrices
    build_dct_kernel<<<nblk, threads, 0, stream>>>(D);
    build_weights_kernel<<<nblk, threads, 0, stream>>>(att, row_w, col_w, Wrow, Wcol);

    // 2) E = D^T * (W * D), for row and col weights — WMMA f32 GEMMs
    dim3 ggrid(NN / 16, NN / 16);
    gemm512_wmma_kernel<false><<<ggrid, 32, 0, stream>>>(Wrow, D, Prow); // P = W*D
    gemm512_wmma_kernel<false><<<ggrid, 32, 0, stream>>>(Wcol, D, Pcol);
    gemm512_wmma_kernel<true ><<<ggrid, 32, 0, stream>>>(D, Prow, Erow); // E = D^T*P
    gemm512_wmma_kernel<true ><<<ggrid, 32, 0, stream>>>(D, Pcol, Ecol);

    // 3) Memory-bound streaming pass over x (192 MiB), per-batch dot products
    dim3 rgrid(WGS_PER_BATCH, NB);
    batch_reduce_kernel<<<rgrid, 256, 0, stream>>>(x, Erow, Ecol, part);

    // 4) Deterministic final reduce + sigmoid
    finalize_kernel<<<NB, 32, 0, stream>>>(part, out);
}